// BatchNormSPDMean_25778393711096
// MI455X (gfx1250) — compile-verified
//
#include <hip/hip_runtime.h>

#define N 64
#define LDM 68              // padded LDS row stride: 68 % 64 = 4 -> conflict-free col reads; rows 16B aligned
#define AUGLD 136           // augmented [A|I] row stride (2 slots wide)
#define SLOT (N * LDM)      // 4352 floats per 64x64 LDS matrix slot
#define NTHREADS 256
#define NS_ITERS 14
#define EXPM_SQUARINGS 6
#define EXPM_TERMS 12

typedef __attribute__((ext_vector_type(2))) float v2f;
typedef __attribute__((ext_vector_type(8))) float v8f;
typedef __attribute__((ext_vector_type(4))) float fvec4;
typedef __attribute__((ext_vector_type(4))) int   v4i;

// Address-space-qualified int4 pointee types for the async builtin
// (clang expects: arg0 = v4i in AS1/global, arg1 = v4i in AS3/LDS).
typedef __attribute__((address_space(1))) v4i glob_v4i;
typedef __attribute__((address_space(3))) v4i lds_v4i;

// CDNA5 async global->LDS path (ASYNCcnt) if the toolchain exposes it.
#if __has_builtin(__builtin_amdgcn_global_load_async_to_lds_b128) && \
    __has_builtin(__builtin_amdgcn_s_wait_asynccnt)
#define HAVE_GFX1250_ASYNC 1
#else
#define HAVE_GFX1250_ASYNC 0
#endif

// ---------------- WMMA f32 16x16x4 based 64x64x64 GEMM on LDS, 8 waves ----------------
// A,B,D are LDS 64x64 matrices with row stride LDM. BT=true computes D = A * B^T.
template <bool BT>
__device__ __forceinline__ void gemm64(const float* A, const float* Bm, float* D, int tid)
{
    const int wave = tid >> 5;
    const int lane = tid & 31;
    const int half = lane >> 4;   // 0: lanes 0-15, 1: lanes 16-31
    const int l16  = lane & 15;
    for (int t = wave; t < 16; t += 8) {          // 2 of the 16 output tiles per wave
        const int m0 = (t >> 2) << 4;
        const int n0 = (t & 3) << 4;
        v8f acc = {0.f, 0.f, 0.f, 0.f, 0.f, 0.f, 0.f, 0.f};
#pragma unroll
        for (int k0 = 0; k0 < N; k0 += 4) {
            const int ka = k0 + (half << 1);      // K = {0,1} lanes 0-15, {2,3} lanes 16-31
            v2f a, b;
            a.x = A[(m0 + l16) * LDM + ka];
            a.y = A[(m0 + l16) * LDM + ka + 1];
            if (!BT) {
                b.x = Bm[(ka    ) * LDM + n0 + l16];
                b.y = Bm[(ka + 1) * LDM + n0 + l16];
            } else {
                b.x = Bm[(n0 + l16) * LDM + ka];
                b.y = Bm[(n0 + l16) * LDM + ka + 1];
            }
            acc = __builtin_amdgcn_wmma_f32_16x16x4_f32(false, a, false, b,
                                                        (short)0, acc, false, false);
        }
#pragma unroll
        for (int r = 0; r < 8; ++r)
            D[(m0 + half * 8 + r) * LDM + n0 + l16] = acc[r];
    }
}

// ---------------- Gauss-Jordan inverse of SPD 64x64 in augmented LDS [A|I] ----------------
// aug: 64 rows x 128 cols, row stride AUGLD. No pivoting (SPD). Column k is never
// revisited after step k, so we skip writing it and avoid the f-read hazard.
__device__ __forceinline__ void invert64(float* aug, int tid)
{
    const int irow = tid >> 2;            // 4 threads per row
    const int j0   = (tid & 3) << 5;      // 32 cols per thread
    for (int k = 0; k < N; ++k) {
        __syncthreads();                  // previous elimination done
        const float pinv = 1.0f / aug[k * AUGLD + k];
        if (tid < 128 && tid != k) aug[k * AUGLD + tid] *= pinv;   // scale pivot row (skip col k)
        __syncthreads();                  // pivot row ready
        if (irow != k) {
            const float f = aug[irow * AUGLD + k];
            for (int j = j0; j < j0 + 32; ++j)
                if (j != k) aug[irow * AUGLD + j] -= f * aug[k * AUGLD + j];
        }
    }
    __syncthreads();
}

// ---------------- small elementwise helpers on LDS 64x64 (stride LDM) ----------------
__device__ __forceinline__ void copyMat(float* D, const float* S, int tid)
{
    for (int e = tid; e < N * N; e += NTHREADS) {
        int r = e >> 6, c = e & 63;
        D[r * LDM + c] = S[r * LDM + c];
    }
}

__device__ __forceinline__ void symmetrize(float* G, int tid)
{
    for (int e = tid; e < N * N; e += NTHREADS) {
        int r = e >> 6, c = e & 63;
        if (r < c) {
            float a = G[r * LDM + c], b = G[c * LDM + r];
            float m = 0.5f * (a + b);
            G[r * LDM + c] = m;
            G[c * LDM + r] = m;
        }
    }
}

// ---------------- coupled Newton-Schulz: Y -> sqrt(M), Z -> isqrt(M) ----------------
// Needs 5 distinct slots (Y,Z,T,SY,SZ); SZ may alias M (M only read during init).
__device__ __forceinline__ void ns_sqrt(const float* M, float*& Y, float*& Z, float* T,
                                        float*& SY, float*& SZ, float* red, int tid)
{
    if (tid < N) red[tid] = M[tid * LDM + tid];
    __syncthreads();
    if (tid == 0) {
        float s = 0.f;
        for (int i = 0; i < N; ++i) s += red[i];
        red[0] = s;
    }
    __syncthreads();
    const float s  = red[0];          // trace >= spectral norm for SPD -> ||M/s|| <= 1
    const float is = 1.0f / s;
    for (int e = tid; e < N * N; e += NTHREADS) {
        int r = e >> 6, c = e & 63;
        Y[r * LDM + c] = M[r * LDM + c] * is;
        Z[r * LDM + c] = (r == c) ? 1.f : 0.f;
    }
    __syncthreads();
    for (int it = 0; it < NS_ITERS; ++it) {
        gemm64<false>(Z, Y, T, tid);                 // T = Z*Y
        __syncthreads();
        for (int e = tid; e < N * N; e += NTHREADS) {// T = 1.5 I - 0.5 T
            int r = e >> 6, c = e & 63;
            float v = -0.5f * T[r * LDM + c];
            if (r == c) v += 1.5f;
            T[r * LDM + c] = v;
        }
        __syncthreads();
        gemm64<false>(Y, T, SY, tid);                // Y' = Y*W
        gemm64<false>(T, Z, SZ, tid);                // Z' = W*Z
        __syncthreads();
        float* tmp;
        tmp = Y; Y = SY; SY = tmp;
        tmp = Z; Z = SZ; SZ = tmp;
    }
    const float rs = sqrtf(s), irs = 1.0f / rs;
    for (int e = tid; e < N * N; e += NTHREADS) {
        int r = e >> 6, c = e & 63;
        Y[r * LDM + c] *= rs;    // sqrt(M)
        Z[r * LDM + c] *= irs;   // isqrt(M)
    }
    __syncthreads();
}

// ================= kernel 0: zero the accumulators =================
__global__ void zero_kernel(float* p, int n)
{
    int i = blockIdx.x * blockDim.x + threadIdx.x;
    if (i < n) p[i] = 0.f;
}

// ================= kernel 1: batch reduction: sum(X) and sum(inv(X)) =================
__global__ void __launch_bounds__(NTHREADS)
reduce_kernel(const float* __restrict__ data, float* wsA, float* wsH, int batch)
{
    __shared__ float aug[N * AUGLD];
    const int tid = threadIdx.x;
    float accA[16], accH[16];
#pragma unroll
    for (int q = 0; q < 16; ++q) { accA[q] = 0.f; accH[q] = 0.f; }
    const int e0 = tid * 16;                       // this thread's 16 flat elements
    for (int m = blockIdx.x; m < batch; m += gridDim.x) {
        const float4* src = (const float4*)(data + (size_t)m * 4096);
#pragma unroll
        for (int q4 = 0; q4 < 4; ++q4) {
            float4 v = src[tid * 4 + q4];
            int e = e0 + q4 * 4;
            int r = e >> 6, c = e & 63;
            *(float4*)&aug[r * AUGLD + c] = v;     // left half = X
            accA[q4 * 4 + 0] += v.x;
            accA[q4 * 4 + 1] += v.y;
            accA[q4 * 4 + 2] += v.z;
            accA[q4 * 4 + 3] += v.w;
#pragma unroll
            for (int j = 0; j < 4; ++j)            // right half = I
                aug[r * AUGLD + 64 + c + j] = (c + j == r) ? 1.f : 0.f;
        }
        __syncthreads();
        invert64(aug, tid);                        // right half = X^-1
#pragma unroll
        for (int q = 0; q < 16; ++q) {
            int e = e0 + q;
            int r = e >> 6, c = e & 63;
            accH[q] += aug[r * AUGLD + 64 + c];
        }
        __syncthreads();                           // before next matrix overwrites aug
    }
#pragma unroll
    for (int q = 0; q < 16; ++q) {
        int e = e0 + q;
        __hip_atomic_fetch_add(&wsA[e], accA[q], __ATOMIC_RELAXED, __HIP_MEMORY_SCOPE_AGENT);
        __hip_atomic_fetch_add(&wsH[e], accH[q], __ATOMIC_RELAXED, __HIP_MEMORY_SCOPE_AGENT);
    }
}

// ================= kernel 2: single-workgroup SPD pipeline -> L = B_s * M_is =================
__global__ void __launch_bounds__(NTHREADS)
solve_kernel(const float* wsA, const float* wsH, const float* __restrict__ cb,
             float* wsL, float batch)
{
    __shared__ float lds[8 * SLOT];               // 8 matrix slots (~139 KB of 320 KB/WGP)
    __shared__ float red[N];
    const int tid = threadIdx.x;
    float* W0 = lds + 0 * SLOT;
    float* W1 = lds + 1 * SLOT;
    float* W2 = lds + 2 * SLOT;
    float* W3 = lds + 3 * SLOT;
    float* W4 = lds + 4 * SLOT;
    float* P1 = lds + 5 * SLOT;
    float* P2 = lds + 6 * SLOT;
    float* P3 = lds + 7 * SLOT;
    const float ib = 1.0f / batch;

    // A -> W0 ; augmented [mean(inv X) | I] spans W1,W2
    float* aug = W1;
    for (int e = tid; e < N * N; e += NTHREADS) {
        int r = e >> 6, c = e & 63;
        W0[r * LDM + c] = wsA[e] * ib;
        aug[r * AUGLD + c] = wsH[e] * ib;
        aug[r * AUGLD + 64 + c] = (r == c) ? 1.f : 0.f;
    }
    __syncthreads();
    invert64(aug, tid);                            // H = inv(mean(inv X))
    for (int e = tid; e < N * N; e += NTHREADS) {
        int r = e >> 6, c = e & 63;
        P1[r * LDM + c] = aug[r * AUGLD + 64 + c];
    }
    __syncthreads();

    // A_s = sqrt(A) -> P2 ; A_is = isqrt(A) -> P3
    {
        float *Y = W1, *Z = W2, *SY = W4, *SZ = W0;
        ns_sqrt(W0, Y, Z, W3, SY, SZ, red, tid);
        copyMat(P2, Y, tid);
        copyMat(P3, Z, tid);
        __syncthreads();
    }
    // C = A_is * H * A_is
    gemm64<false>(P3, P1, W0, tid); __syncthreads();
    gemm64<false>(W0, P3, W1, tid); __syncthreads();
    // C_s = sqrt(C) -> P1
    {
        float *Y = W0, *Z = W2, *SY = W4, *SZ = W1;
        ns_sqrt(W1, Y, Z, W3, SY, SZ, red, tid);
        copyMat(P1, Y, tid);
        __syncthreads();
    }
    // G = sym(A_s * C_s * A_s) -> W1
    gemm64<false>(P2, P1, W0, tid); __syncthreads();
    gemm64<false>(W0, P2, W1, tid); __syncthreads();
    symmetrize(W1, tid); __syncthreads();
    // M_is = isqrt(G) -> P3
    {
        float *Y = W0, *Z = W2, *SY = W4, *SZ = W1;
        ns_sqrt(W1, Y, Z, W3, SY, SZ, red, tid);
        copyMat(P3, Z, tid);
        __syncthreads();
    }
    // B = expm(sym(cb)) via scaling-and-squaring with Horner Taylor
    float* E = W0;
    float* R = W1;
    float* T = W2;
    const float sc = 1.0f / (float)(1 << EXPM_SQUARINGS);
    for (int e = tid; e < N * N; e += NTHREADS) {
        int r = e >> 6, c = e & 63;
        E[r * LDM + c] = 0.5f * (cb[r * 64 + c] + cb[c * 64 + r]) * sc;
        R[r * LDM + c] = (r == c) ? 1.f : 0.f;
    }
    __syncthreads();
    for (int k = EXPM_TERMS; k >= 1; --k) {        // R = I + (E*R)/k
        gemm64<false>(E, R, T, tid); __syncthreads();
        const float invk = 1.0f / (float)k;
        for (int e = tid; e < N * N; e += NTHREADS) {
            int r = e >> 6, c = e & 63;
            float v = T[r * LDM + c] * invk;
            if (r == c) v += 1.f;
            R[r * LDM + c] = v;
        }
        __syncthreads();
    }
    for (int q = 0; q < EXPM_SQUARINGS; ++q) {     // R = R*R, repeated
        gemm64<false>(R, R, T, tid); __syncthreads();
        float* tmp = R; R = T; T = tmp;
    }
    // B_s = sqrt(B); L = B_s * M_is -> wsL
    {
        float *Y = W0, *Z = W3, *SY = T, *SZ = R;  // SZ aliases B (allowed)
        ns_sqrt(R, Y, Z, W4, SY, SZ, red, tid);
        gemm64<false>(Y, P3, SY, tid); __syncthreads();
        for (int e = tid; e < N * N; e += NTHREADS) {
            int r = e >> 6, c = e & 63;
            wsL[e] = SY[r * LDM + c];
        }
    }
}

// ---------------- async global->LDS load of one 64x64 fp32 matrix (4 b128 per wave) ----------------
__device__ __forceinline__ void issue_async_load_mat(const float* src, float* buf, int tid)
{
#if HAVE_GFX1250_ASYNC
#pragma unroll
    for (int q = 0; q < 4; ++q) {
        int cq = q * NTHREADS + tid;               // coalesced 16B chunk index
        int e = cq * 4;
        int r = e >> 6, c = e & 63;
        __builtin_amdgcn_global_load_async_to_lds_b128(
            (glob_v4i*)(__UINTPTR_TYPE__)(src + (size_t)cq * 4),
            (lds_v4i*)(unsigned)(__UINTPTR_TYPE__)&buf[r * LDM + c],
            0, 0);
    }
#else
    (void)src; (void)buf; (void)tid;
#endif
}

// ================= kernel 3: streaming congruence out[i] = L * X_i * L^T =================
// Double-buffered: async b128 loads write LDS directly (ASYNCcnt); s_wait_asynccnt 4
// retires the current buffer (loads complete in order, 4 issued per wave per buffer).
__global__ void __launch_bounds__(NTHREADS)
apply_kernel(const float* __restrict__ data, const float* __restrict__ wsL,
             float* __restrict__ out, int batch)
{
    __shared__ float Lb[SLOT];
    __shared__ float Xb0[SLOT];
    __shared__ float Xb1[SLOT];
    __shared__ float Tb[SLOT];
    const int tid = threadIdx.x;
    for (int e = tid; e < N * N; e += NTHREADS) {
        int r = e >> 6, c = e & 63;
        Lb[r * LDM + c] = wsL[e];
    }
#if HAVE_GFX1250_ASYNC
    if (blockIdx.x < (unsigned)batch)              // prologue: prime buffer 0
        issue_async_load_mat(data + (size_t)blockIdx.x * 4096, Xb0, tid);
#endif
    __syncthreads();
    int p = 0;
    for (int m = blockIdx.x; m < batch; m += gridDim.x) {
        float* Xc = p ? Xb1 : Xb0;                 // current buffer
        float* Xn = p ? Xb0 : Xb1;                 // next buffer
        const int mn = m + (int)gridDim.x;
        const bool hn = mn < batch;                // uniform across the block
#if HAVE_GFX1250_ASYNC
        if (hn) issue_async_load_mat(data + (size_t)mn * 4096, Xn, tid);
        if (hn) __builtin_amdgcn_s_wait_asynccnt(4);   // current buffer's 4 loads retired
        else    __builtin_amdgcn_s_wait_asynccnt(0);
        __syncthreads();
#else
        {
            const float4* src = (const float4*)(data + (size_t)m * 4096);
#pragma unroll
            for (int q = 0; q < 4; ++q) {
                int cq = q * NTHREADS + tid;
                float4 v = src[cq];
                int e = cq * 4;
                int r = e >> 6, c = e & 63;
                *(float4*)&Xc[r * LDM + c] = v;
            }
            if (hn) __builtin_prefetch(data + (size_t)mn * 4096 + tid * 16, 0, 1);
            __syncthreads();
            (void)Xn;
        }
#endif
        gemm64<false>(Lb, Xc, Tb, tid);            // T = L * X
        __syncthreads();
        gemm64<true>(Tb, Lb, Xc, tid);             // O = T * L^T (reuse X buffer)
        __syncthreads();
        fvec4* dst = (fvec4*)(out + (size_t)m * 4096);
#pragma unroll
        for (int q = 0; q < 4; ++q) {              // NT stores: keep `data` resident in L2
            int cq = q * NTHREADS + tid;
            int e = cq * 4;
            int r = e >> 6, c = e & 63;
            fvec4 v = *(const fvec4*)&Xc[r * LDM + c];
            __builtin_nontemporal_store(v, dst + cq);
        }
        __syncthreads();                           // LDS reads done before next reuse
        p ^= 1;
    }
}

// ================= host-side launcher =================
extern "C" void kernel_launch(void* const* d_in, const int* in_sizes, int n_in,
                              void* d_out, int out_size, void* d_ws, size_t ws_size,
                              hipStream_t stream)
{
    const float* data = (const float*)d_in[0];   // (BATCH, 64, 64) f32
    const float* cb   = (const float*)d_in[1];   // (64, 64) f32
    float* out = (float*)d_out;
    float* ws  = (float*)d_ws;
    const int batch = in_sizes[0] / (N * N);

    float* wsA = ws;            // 4096 floats: sum(X)
    float* wsH = ws + 4096;     // 4096 floats: sum(inv X)
    float* wsL = ws + 8192;     // 4096 floats: combined transform L

    hipLaunchKernelGGL(zero_kernel, dim3(32), dim3(NTHREADS), 0, stream, wsA, 8192);
    hipLaunchKernelGGL(reduce_kernel, dim3(512), dim3(NTHREADS), 0, stream,
                       data, wsA, wsH, batch);
    hipLaunchKernelGGL(solve_kernel, dim3(1), dim3(NTHREADS), 0, stream,
                       wsA, wsH, cb, wsL, (float)batch);
    hipLaunchKernelGGL(apply_kernel, dim3(2048), dim3(NTHREADS), 0, stream,
                       data, wsL, out, batch);
}